// MHALayer_75196287418866
// MI455X (gfx1250) — compile-verified
//
#include <hip/hip_runtime.h>
#include <hip/hip_bf16.h>
#include <cstdint>

// ---------------------------------------------------------------------------
// MHA decode (1 query, N=16384 keys) for MI455X / gfx1250.
// Memory-bound: 536 MB of K/V @ 23.3 TB/s => ~23 us floor. Strategy:
//   k1: flash-decode partials per (batch, 256-key chunk); K staged to LDS via
//       async b128 copies (ASYNCcnt); QK as dense f32 WMMA GEMM against a
//       block-diagonal Q matrix; V consumed with coalesced non-temporal b32.
//   k2a: cross-chunk softmax combine.
//   k2b: 32x128x128 output projection as f32 WMMA GEMM.
// ---------------------------------------------------------------------------

#define N_HEADS 8
#define EMBED   128
#define BATCH   32
#define NKEYS   16384
#define CH      256            // keys per chunk
#define CHUNKS  (NKEYS / CH)   // 64
#define KROW    132            // padded LDS row stride (floats) -> conflict-free A frags
#define PROW    (CH + 4)       // padded p_s row stride (floats)
#define THREADS 256
#define WSBLK   144            // floats per (batch,chunk) partial: m[8] l[8] acc[128]
#define LOG2E   1.4426950408889634f

typedef __attribute__((ext_vector_type(2))) float v2f;
typedef __attribute__((ext_vector_type(8))) float v8f;
typedef int v4i __attribute__((vector_size(16)));
typedef v4i __attribute__((address_space(1))) *g_v4i_ptr;  // global ptr
typedef v4i __attribute__((address_space(3))) *l_v4i_ptr;  // LDS ptr

// ---- f32 WMMA 16x16x4 ------------------------------------------------------
__device__ __forceinline__ v8f wmma_f32_16x16x4(v2f a, v2f b, v8f c) {
#if __has_builtin(__builtin_amdgcn_wmma_f32_16x16x4_f32)
  // (neg_a, A, neg_b, B, c_mod, C, reuse_a, reuse_b)
  return __builtin_amdgcn_wmma_f32_16x16x4_f32(false, a, false, b, (short)0, c,
                                               false, false);
#else
  #pragma unroll
  for (int i = 0; i < 8; ++i) c[i] += a.x * b.x + a.y * b.y;
  return c;
#endif
}

// ---- async global->LDS 16B copy (proved present by round-1 diagnostics) ----
__device__ __forceinline__ void async_copy16(const float* g, float* l) {
#if __has_builtin(__builtin_amdgcn_global_load_async_to_lds_b128)
  __builtin_amdgcn_global_load_async_to_lds_b128(
      (g_v4i_ptr)g, (l_v4i_ptr)l, /*offset=*/0, /*cpol=*/0);
#else
  *(float4*)l = *(const float4*)g;
#endif
}

__device__ __forceinline__ void wait_async() {
#if __has_builtin(__builtin_amdgcn_global_load_async_to_lds_b128)
#if __has_builtin(__builtin_amdgcn_s_wait_asynccnt)
  __builtin_amdgcn_s_wait_asynccnt(0);
#else
  asm volatile("s_wait_asynccnt 0" ::: "memory");
#endif
#endif
}

// ===========================================================================
// Kernel 1: per (batch, chunk) attention partials
// ===========================================================================
__global__ void __launch_bounds__(THREADS)
attn_partial_kernel(const float* __restrict__ query,
                    const float* __restrict__ keys,
                    const float* __restrict__ values,
                    const float* __restrict__ mask,
                    float* __restrict__ ws1) {
  extern __shared__ float smem[];
  float* K_s    = smem;                    // CH*KROW
  float* p_s    = K_s + CH * KROW;         // 8*PROW (logits -> exp values)
  float* q_s    = p_s + 8 * PROW;          // 128
  float* mask_s = q_s + EMBED;             // CH
  float* red    = mask_s + CH;             // 256
  float* av     = red + THREADS;           // 256
  float* hstat  = av + THREADS;            // 16: m[8], l[8]

  const int t     = threadIdx.x;
  const int blk   = blockIdx.x;
  const int b     = blk / CHUNKS;
  const int chunk = blk % CHUNKS;
  const int n0    = chunk * CH;

  if (t < EMBED) q_s[t] = query[b * EMBED + t];
  mask_s[t] = mask[(size_t)b * NKEYS + n0 + t];

  // Async-stage K chunk (256 rows x 512B), padded rows. 32 iters x 4KB.
  {
    const float* gbase = keys + (size_t)(b * NKEYS + n0) * EMBED;
    #pragma unroll 4
    for (int it = 0; it < (CH * EMBED) / (THREADS * 4); ++it) {
      int lin = it * THREADS + t;      // 16B-chunk index
      int row = lin >> 5;              // 32 x 16B per key row
      int c4  = (lin & 31) << 2;       // float offset within row
      async_copy16(gbase + (size_t)row * EMBED + c4, K_s + row * KROW + c4);
    }
  }
  wait_async();
  __syncthreads();

  const int wv  = t >> 5;     // wave 0..7 -> 32 keys each
  const int L   = t & 31;     // lane
  const int lh  = L & 15;     // column (head for lh<8) / A row
  const int khi = L >> 4;     // 0/1 -> K sub-pair selector

  // B fragments: block-diagonal Q, B[k, h] = q[k] iff h == k/16 (h<8).
  // Loads are unconditional (uniform LDS broadcast); the lane-dependent
  // predicate becomes v_cndmask instead of exec-masked loads.
  v2f bfrag[32];
  #pragma unroll
  for (int kk = 0; kk < 32; ++kk) {
    const int kx = 4 * kk + 2 * khi;
    const float qx = q_s[kx];
    const float qy = q_s[kx + 1];
    bfrag[kk].x = ((kx >> 4) == lh) ? qx : 0.0f;
    bfrag[kk].y = (((kx + 1) >> 4) == lh) ? qy : 0.0f;
  }

  // QK: two 16-key tiles per wave, K=128 in 32 WMMA steps each.
  #pragma unroll
  for (int tile = 0; tile < 2; ++tile) {
    const int keyBase = wv * 32 + tile * 16;
    const float* arow = K_s + (keyBase + lh) * KROW + 2 * khi;
    v8f acc = {};
    #pragma unroll
    for (int kk = 0; kk < 32; ++kk) {
      v2f a = *(const v2f*)(arow + 4 * kk);     // ds_load_b64, conflict-free
      acc = wmma_f32_16x16x4(a, bfrag[kk], acc);
    }
    // D layout: vgpr r, lane L -> [key = r + 8*(L/16), col = L%16]
    if (lh < N_HEADS) {
      #pragma unroll
      for (int r = 0; r < 8; ++r) {
        int key = keyBase + r + 8 * khi;
        p_s[lh * PROW + key] = acc[r] * 0.25f + mask_s[key];
      }
    }
  }
  __syncthreads();

  // Per-head chunk max.
  {
    int h = t & 7, j = t >> 3;
    float mx = -3.0e38f;
    #pragma unroll
    for (int i = 0; i < 8; ++i) mx = fmaxf(mx, p_s[h * PROW + j * 8 + i]);
    red[t] = mx;
  }
  __syncthreads();
  if (t < 8) {
    float mx = -3.0e38f;
    for (int j = 0; j < 32; ++j) mx = fmaxf(mx, red[t + 8 * j]);
    hstat[t] = mx;
  }
  __syncthreads();

  // exp + per-head sum.
  {
    int h = t & 7, j = t >> 3;
    float M = hstat[h], s = 0.f;
    #pragma unroll
    for (int i = 0; i < 8; ++i) {
      float e = exp2f((p_s[h * PROW + j * 8 + i] - M) * LOG2E);
      p_s[h * PROW + j * 8 + i] = e;
      s += e;
    }
    red[t] = s;
  }
  __syncthreads();
  if (t < 8) {
    float s = 0.f;
    for (int j = 0; j < 32; ++j) s += red[t + 8 * j];
    hstat[8 + t] = s;
  }
  __syncthreads();

  // AV: thread = (concat dim c, key half); V reads are 128B-contiguous per
  // wave instruction, streamed non-temporally (each byte used exactly once).
  {
    int c = t & 127, half = t >> 7, hc = c >> 4;
    const float* vb = values + (size_t)(b * NKEYS + n0 + half * 128) * EMBED + c;
    const float* pr = p_s + hc * PROW + half * 128;
    float a0 = 0.f, a1 = 0.f;
    #pragma unroll 8
    for (int j = 0; j < 128; j += 2) {
      float v0 = __builtin_nontemporal_load(vb + (size_t)j * EMBED);
      float v1 = __builtin_nontemporal_load(vb + (size_t)(j + 1) * EMBED);
      a0 += pr[j] * v0;
      a1 += pr[j + 1] * v1;
    }
    av[t] = a0 + a1;
  }
  __syncthreads();

  float* out = ws1 + (size_t)blk * WSBLK;
  if (t < 128)       out[16 + t] = av[t] + av[t + 128];
  else if (t < 144)  out[t - 128] = hstat[t - 128];   // m[0..7], l[0..7]
}

// ===========================================================================
// Kernel 2a: combine chunk partials -> heads_concat (B x 128)
// ===========================================================================
__global__ void __launch_bounds__(128)
combine_kernel(const float* __restrict__ ws1, float* __restrict__ heads) {
  const int b = blockIdx.x, t = threadIdx.x;  // t = concat dim
  const int h = t >> 4;
  const float* base = ws1 + (size_t)b * CHUNKS * WSBLK;
  float M = -3.0e38f;
  for (int ch = 0; ch < CHUNKS; ++ch) M = fmaxf(M, base[ch * WSBLK + h]);
  float S = 0.f, A = 0.f;
  for (int ch = 0; ch < CHUNKS; ++ch) {
    const float* p = base + ch * WSBLK;
    float sc = exp2f((p[h] - M) * LOG2E);
    S += p[8 + h] * sc;
    A += p[16 + t] * sc;
  }
  heads[b * EMBED + t] = A / S;
}

// ===========================================================================
// Kernel 2b: out(32x128) = heads(32x128) @ W_out(128x128) via f32 WMMA
// ===========================================================================
__global__ void __launch_bounds__(THREADS)
proj_kernel(const float* __restrict__ heads, const float* __restrict__ Wout,
            float* __restrict__ out) {
  const int t = threadIdx.x;
  const int wv = t >> 5;            // N-tile 0..7
  const int L = t & 31, lh = L & 15, khi = L >> 4;
  #pragma unroll
  for (int mt = 0; mt < 2; ++mt) {  // M-tiles (batch rows 0..15, 16..31)
    v8f acc = {};
    #pragma unroll
    for (int kk = 0; kk < 32; ++kk) {
      int k = 4 * kk + 2 * khi;
      v2f a = *(const v2f*)(heads + (mt * 16 + lh) * EMBED + k);
      v2f bb;
      bb.x = Wout[(size_t)k * EMBED + wv * 16 + lh];
      bb.y = Wout[(size_t)(k + 1) * EMBED + wv * 16 + lh];
      acc = wmma_f32_16x16x4(a, bb, acc);
    }
    #pragma unroll
    for (int r = 0; r < 8; ++r) {
      int row = mt * 16 + r + 8 * khi;
      out[(size_t)row * EMBED + wv * 16 + lh] = acc[r];
    }
  }
}

// ===========================================================================
extern "C" void kernel_launch(void* const* d_in, const int* in_sizes, int n_in,
                              void* d_out, int out_size, void* d_ws,
                              size_t ws_size, hipStream_t stream) {
  const float* query  = (const float*)d_in[0];
  const float* keys   = (const float*)d_in[1];
  const float* values = (const float*)d_in[2];
  const float* mask   = (const float*)d_in[3];
  const float* Wout   = (const float*)d_in[4];
  float* out = (float*)d_out;

  float* ws1   = (float*)d_ws;                          // 2048*144 floats
  float* heads = ws1 + (size_t)BATCH * CHUNKS * WSBLK;  // +32*128 floats

  const size_t smem =
      (size_t)(CH * KROW + 8 * PROW + EMBED + CH + THREADS + THREADS + 16) *
      sizeof(float);  // ~147 KB dynamic LDS (gfx1250 WGP has 320 KB)

  attn_partial_kernel<<<BATCH * CHUNKS, THREADS, smem, stream>>>(
      query, keys, values, mask, ws1);
  combine_kernel<<<BATCH, 128, 0, stream>>>(ws1, heads);
  proj_kernel<<<1, THREADS, 0, stream>>>(heads, Wout, out);
}